// LossFunction_12532714569881
// MI455X (gfx1250) — compile-verified
//
#include <hip/hip_runtime.h>
#include <math.h>

typedef __attribute__((ext_vector_type(2))) float v2f;
typedef __attribute__((ext_vector_type(4))) float v4f;
typedef __attribute__((ext_vector_type(8))) float v8f;

#define NROWS 8192
#define DDIM 256
#define ROWSTRIDE 512   // 2*D floats between consecutive n indices
#define EPSV 1e-8f

// ---------------------------------------------------------------------------
// Kernel 1: per-row L2 norms of positive (x[:,0,:]) and anchor (x[:,1,:]),
// plus the per-row dot product dot(P_i, A_i) (the diagonal of P @ A^T).
// One wave (32 lanes) per row; 8 floats per lane via two float4 loads.
// ---------------------------------------------------------------------------
__global__ __launch_bounds__(256) void norms_kernel(const float* __restrict__ x,
                                                    float* __restrict__ pn,
                                                    float* __restrict__ an,
                                                    float* __restrict__ pd) {
  const int wave = threadIdx.x >> 5;
  const int lane = threadIdx.x & 31;
  const int row  = blockIdx.x * 8 + wave;
  const float* prow = x + (size_t)row * ROWSTRIDE;        // positive row
  const float* arow = prow + DDIM;                        // anchor row
  float sp = 0.f, sa = 0.f, sd = 0.f;
#pragma unroll
  for (int i = 0; i < 2; ++i) {
    v4f p = *(const v4f*)(prow + lane * 8 + i * 4);
    v4f a = *(const v4f*)(arow + lane * 8 + i * 4);
#pragma unroll
    for (int j = 0; j < 4; ++j) {
      sp += p[j] * p[j];
      sa += a[j] * a[j];
      sd += p[j] * a[j];
    }
  }
#pragma unroll
  for (int m = 16; m >= 1; m >>= 1) {
    sp += __shfl_xor(sp, m, 32);
    sa += __shfl_xor(sa, m, 32);
    sd += __shfl_xor(sd, m, 32);
  }
  if (lane == 0) { pn[row] = sqrtf(sp); an[row] = sqrtf(sa); pd[row] = sd; }
}

// ---------------------------------------------------------------------------
// Kernel 2: fused fp32 WMMA GEMM (P @ A^T, K=256) + fixed-shift sum-of-exp
// per row.  Block = 32-row strip; its 8 waves split the 8192 columns
// (1024 each).  Wave tile = 32x32: 2 A frags x 2 B frags -> 4 WMMAs per
// 4 b64 loads; each B element reused across 32 rows.
// Epilogue is branch-free (diagonal handled from pd[] in the finalizer).
// ---------------------------------------------------------------------------
__global__ __launch_bounds__(256) void gemm_lse_kernel(const float* __restrict__ x,
                                                       const float* __restrict__ pn,
                                                       const float* __restrict__ an,
                                                       const float* __restrict__ pd,
                                                       const float* __restrict__ wp,
                                                       const float* __restrict__ bp,
                                                       float* __restrict__ rowloss) {
  __shared__ float s_part[8][32];

  const int tid  = threadIdx.x;
  const int wave = tid >> 5;
  const int lane = tid & 31;
  const int half = lane >> 4;     // 0: rows v / K0-K1   1: rows v+8 / K2-K3
  const int mrow = lane & 15;
  const int r0   = blockIdx.x * 32;

  const float Wv    = wp[0];
  const float Bv    = bp[0];
  const float shift = fabsf(Wv) + Bv;   // max possible logit (|cos| <= 1)
  const float Bs    = Bv - shift;       // folded exp-argument offset

  // A-fragment sources: positive rows for the two 16-row strips
  const float* Pbase0 = x + (size_t)(r0      + mrow) * ROWSTRIDE;
  const float* Pbase1 = x + (size_t)(r0 + 16 + mrow) * ROWSTRIDE;

  float pnv[2][8];
#pragma unroll
  for (int st = 0; st < 2; ++st)
#pragma unroll
    for (int v = 0; v < 8; ++v) pnv[st][v] = pn[r0 + 16 * st + v + 8 * half];

  float ssum[2][8];
#pragma unroll
  for (int st = 0; st < 2; ++st)
#pragma unroll
    for (int v = 0; v < 8; ++v) ssum[st][v] = 0.f;

  const int c0 = wave * 1024;
  for (int chunk = 0; chunk < 32; ++chunk) {
    const int cbase = c0 + chunk * 32;
    v8f acc00 = {}, acc01 = {}, acc10 = {}, acc11 = {};
    // B-fragment source: anchor rows (cbase + 16*subcol + mrow)
    const float* B0 = x + (size_t)(cbase + mrow) * ROWSTRIDE + DDIM;

#pragma unroll 4
    for (int k0 = 0; k0 < DDIM; k0 += 4) {
      const int koff = k0 + 2 * half;
      v2f a0 = *(const v2f*)(Pbase0 + koff);
      v2f a1 = *(const v2f*)(Pbase1 + koff);
      v2f b0 = *(const v2f*)(B0 + koff);
      v2f b1 = *(const v2f*)(B0 + 16 * ROWSTRIDE + koff);
      acc00 = __builtin_amdgcn_wmma_f32_16x16x4_f32(false, a0, false, b0, (short)0, acc00, false, false);
      acc01 = __builtin_amdgcn_wmma_f32_16x16x4_f32(false, a0, false, b1, (short)0, acc01, false, false);
      acc10 = __builtin_amdgcn_wmma_f32_16x16x4_f32(false, a1, false, b0, (short)0, acc10, false, false);
      acc11 = __builtin_amdgcn_wmma_f32_16x16x4_f32(false, a1, false, b1, (short)0, acc11, false, false);
    }

    // Epilogue: C layout -> VGPR v holds row (16*strip + v + 8*half),
    // column (cbase + 16*subcol + mrow).  Branch-free.
    float amv[2];
#pragma unroll
    for (int sc = 0; sc < 2; ++sc) amv[sc] = an[cbase + 16 * sc + mrow];

#define EPILOGUE(ACC, ST, SC)                                                 \
    {                                                                         \
      const float am = amv[SC];                                               \
      _Pragma("unroll")                                                       \
      for (int v = 0; v < 8; ++v) {                                           \
        float denom = fmaxf(pnv[ST][v] * am, EPSV);                           \
        float scale = Wv * __builtin_amdgcn_rcpf(denom);                      \
        float arg   = (ACC)[v] * scale + Bs;   /* logit - shift */            \
        ssum[ST][v] += __expf(arg);                                           \
      }                                                                       \
    }
    EPILOGUE(acc00, 0, 0)
    EPILOGUE(acc01, 0, 1)
    EPILOGUE(acc10, 1, 0)
    EPILOGUE(acc11, 1, 1)
#undef EPILOGUE
  }

  // Reduce partial exp-sums across the 16 lanes that share each row
  // (xor masks 1,2,4,8 stay within a 16-lane half of the wave32).
#pragma unroll
  for (int st = 0; st < 2; ++st) {
#pragma unroll
    for (int v = 0; v < 8; ++v) {
      float s = ssum[st][v];
      s += __shfl_xor(s, 1, 32);
      s += __shfl_xor(s, 2, 32);
      s += __shfl_xor(s, 4, 32);
      s += __shfl_xor(s, 8, 32);
      if (mrow == 0) s_part[wave][16 * st + v + 8 * half] = s;
    }
  }
  __syncthreads();

  if (tid < 32) {
    const int row = r0 + tid;
    float tot = 0.f;
#pragma unroll
    for (int w = 0; w < 8; ++w) tot += s_part[w][tid];
    // Diagonal logit from the precomputed per-row dot product.
    float denom  = fmaxf(pn[row] * an[row], EPSV);
    float dlogit = pd[row] * (Wv * __builtin_amdgcn_rcpf(denom)) + Bv;
    // -logp[i,i] = shift + log(sum_j exp(logit_ij - shift)) - logit_ii
    rowloss[row] = shift + logf(tot) - dlogit;
  }
}

// ---------------------------------------------------------------------------
// Kernel 3: mean over 8192 per-row losses -> scalar.
// ---------------------------------------------------------------------------
__global__ __launch_bounds__(256) void reduce_kernel(const float* __restrict__ rowloss,
                                                     float* __restrict__ out) {
  __shared__ float sm[256];
  float s = 0.f;
  for (int i = threadIdx.x; i < NROWS; i += 256) s += rowloss[i];
  sm[threadIdx.x] = s;
  __syncthreads();
  for (int off = 128; off > 0; off >>= 1) {
    if (threadIdx.x < off) sm[threadIdx.x] += sm[threadIdx.x + off];
    __syncthreads();
  }
  if (threadIdx.x == 0) out[0] = sm[0] / (float)NROWS;
}

extern "C" void kernel_launch(void* const* d_in, const int* in_sizes, int n_in,
                              void* d_out, int out_size, void* d_ws, size_t ws_size,
                              hipStream_t stream) {
  const float* x  = (const float*)d_in[0];
  const float* wp = (const float*)d_in[1];
  const float* bp = (const float*)d_in[2];
  (void)in_sizes; (void)n_in; (void)out_size; (void)ws_size;

  float* pn      = (float*)d_ws;          // 8192 floats
  float* an      = pn + NROWS;            // 8192 floats
  float* pd      = an + NROWS;            // 8192 floats
  float* rowloss = pd + NROWS;            // 8192 floats

  norms_kernel<<<NROWS / 8, 256, 0, stream>>>(x, pn, an, pd);
  gemm_lse_kernel<<<NROWS / 32, 256, 0, stream>>>(x, pn, an, pd, wp, bp, rowloss);
  reduce_kernel<<<1, 256, 0, stream>>>(rowloss, (float*)d_out);
}